// Criterion4OL_38732015075812
// MI455X (gfx1250) — compile-verified
//
#include <hip/hip_runtime.h>
#include <hip/hip_bf16.h>

typedef float v2f __attribute__((ext_vector_type(2)));
typedef float v8f __attribute__((ext_vector_type(8)));

#define NPRED   4096
#define DPRED   78
#define NPTS    72
#define MAXL    8
#define NIMG    192            // S(3) * B(64) per branch
#define BDIM    64
#define HALFW   (15.0f/800.0f)

// ---- dynamic LDS layout for kernel 1 (bytes) ----
#define COST_OFF 0             // float cost[8][4096]      131072
#define S1_OFF   131072        // float s1[4096]            16384
#define GTX_OFF  147456        // float gtx[8][72]           2304
#define GTR_OFF  149760        // float gtreg[8][4]           128
#define GM_OFF   149888        // int   gmask[8]               32
#define RT_OFF   149920        // int   rowtop[8][4]          128
#define MG_OFF   150048        // int   mgmm[4096]          16384
#define MC_OFF   166432        // float mergec[8][128]       4096
#define MI_OFF   170528        // int   mergei[8][128]       4096
#define RED_OFF  174624        // float red[768]             3072
#define LDS_TOTAL 177696

// ===================================================================
// Kernel 1: one block per (branch, s, b) image. Computes the full
// assignment + per-image losses; writes cls_vec[n] to workspace and
// per-image reg/iou scalars.
// ===================================================================
extern "C" __global__ __launch_bounds__(256)
void crit_image_kernel(const float* __restrict__ pred_fir,
                       const float* __restrict__ pred_sec,
                       const float* __restrict__ gt,
                       float* __restrict__ cls_part,   // [2*192][4096]
                       float* __restrict__ reg_part,   // [2*192]
                       float* __restrict__ iou_part)   // [2*192]
{
    extern __shared__ char smem[];
    float* costL = (float*)(smem + COST_OFF);
    float* s1L   = (float*)(smem + S1_OFF);
    float* gtxL  = (float*)(smem + GTX_OFF);
    float* gtrL  = (float*)(smem + GTR_OFF);
    int*   gmL   = (int*)  (smem + GM_OFF);
    int*   rtL   = (int*)  (smem + RT_OFF);
    int*   mgmmL = (int*)  (smem + MG_OFF);
    float* mcL   = (float*)(smem + MC_OFF);
    int*   miL   = (int*)  (smem + MI_OFF);
    float* redL  = (float*)(smem + RED_OFF);

    const int img    = blockIdx.x;        // 0..383
    const int branch = img / NIMG;        // 0 = fir, 1 = sec
    const int sb     = img % NIMG;        // s*64 + b
    const int b      = sb % BDIM;
    const float* pred = (branch ? pred_sec : pred_fir) + (size_t)sb * NPRED * DPRED;
    const float* gtb  = gt + (size_t)b * MAXL * DPRED;

    const int tid = threadIdx.x;

    // ---- 1. GT lanes into LDS ----
    for (int i = tid; i < MAXL * NPTS; i += 256) {
        int l = i / NPTS, k = i % NPTS;
        gtxL[i] = gtb[l * DPRED + 6 + k] * (1.0f / 800.0f);     // gt_x / IMG_W
    }
    for (int i = tid; i < MAXL * 4; i += 256) {
        int l = i / 4, d = i % 4;
        gtrL[i] = gtb[l * DPRED + 2 + d];
    }
    if (tid < MAXL) gmL[tid] = (gtb[tid * DPRED + 1] > 0.5f) ? 1 : 0;

    // ---- 2. softmax[:,1] prepass + match-table init ----
    for (int n = tid; n < NPRED; n += 256) {
        float l0 = pred[(size_t)n * DPRED + 0];
        float l1 = pred[(size_t)n * DPRED + 1];
        s1L[n] = 1.0f / (1.0f + expf(l0 - l1));   // softmax(pred[:,:2])[:,1]
        mgmmL[n] = 0;
    }
    __syncthreads();

    // ---- 3. cost matrix via WMMA f32 16x16x4 ones-reduction ----
    // A(16x4) = abs-diff tile for 16 preds x 4 K-values; B = ones(4x16)
    // => D row m = sum_k |pred_x - gt_x| replicated across all 16 columns.
    const int lane  = tid & 31;
    const int wave  = tid >> 5;
    const int m16   = lane & 15;
    const int khalf = lane >> 4;             // A layout: VGPR0=K(0|2), VGPR1=K(1|3)

    v2f ones; ones[0] = 1.0f; ones[1] = 1.0f;

    for (int t = wave; t < NPRED / 16; t += 8) {      // 32 tiles per wave, uniform
        const int   n    = t * 16 + m16;
        const float* prow = pred + (size_t)n * DPRED + 6;

        // prefetch next tile's pred row (wave-uniform condition; lowers to
        // global_prefetch_b8, no counter use) to hide HBM latency behind WMMAs
        if (t + 8 < NPRED / 16) {
            const float* nprow = pred + (size_t)(n + 128) * DPRED + 6;
            __builtin_prefetch(nprow, 0, 3);
            __builtin_prefetch(nprow + 36, 0, 3);
        }

        v8f acc[MAXL];
        #pragma unroll
        for (int l = 0; l < MAXL; ++l)
            acc[l] = (v8f){0.f,0.f,0.f,0.f,0.f,0.f,0.f,0.f};

        for (int c = 0; c < NPTS / 4; ++c) {          // 18 K-chunks
            const int kb = c * 4 + khalf * 2;
            float px = prow[kb];
            float py = prow[kb + 1];
            // batch all 8 DS loads first so one wait covers them
            float gx[MAXL], gy[MAXL];
            #pragma unroll
            for (int l = 0; l < MAXL; ++l) {
                gx[l] = gtxL[l * NPTS + kb];
                gy[l] = gtxL[l * NPTS + kb + 1];
            }
            #pragma unroll
            for (int l = 0; l < MAXL; ++l) {
                v2f a; a[0] = fabsf(px - gx[l]); a[1] = fabsf(py - gy[l]);
                acc[l] = __builtin_amdgcn_wmma_f32_16x16x4_f32(
                    false, a, false, ones, (short)0, acc[l], false, false);
            }
        }
        // D layout: lane 0 holds M=0..7 (one per VGPR), lane 16 holds M=8..15.
        if (m16 == 0) {
            const int mb = t * 16 + khalf * 8;
            #pragma unroll
            for (int l = 0; l < MAXL; ++l) {
                #pragma unroll
                for (int r = 0; r < 8; ++r) {
                    costL[l * NPRED + mb + r] =
                        acc[l][r] * (1.0f / 72.0f) + (1.0f - s1L[mb + r]);
                }
            }
        }
    }
    __syncthreads();

    // ---- 4. top-4 smallest cost per GT row (wave w owns row w) ----
    {
        const int l = wave;
        float bc0 = 1e30f, bc1 = 1e30f, bc2 = 1e30f, bc3 = 1e30f;
        int   bi0 = -1,    bi1 = -1,    bi2 = -1,    bi3 = -1;
        if (gmL[l]) {
            for (int n = lane; n < NPRED; n += 32) {
                float c = costL[l * NPRED + n];
                if (c < bc3) {                 // ascending n: strict '<' keeps lowest idx on ties
                    if (c < bc2) {
                        bc3 = bc2; bi3 = bi2;
                        if (c < bc1) {
                            bc2 = bc1; bi2 = bi1;
                            if (c < bc0) { bc1 = bc0; bi1 = bi0; bc0 = c; bi0 = n; }
                            else         { bc1 = c;  bi1 = n; }
                        } else { bc2 = c; bi2 = n; }
                    } else { bc3 = c; bi3 = n; }
                }
            }
        }
        mcL[l * 128 + lane * 4 + 0] = bc0; miL[l * 128 + lane * 4 + 0] = bi0;
        mcL[l * 128 + lane * 4 + 1] = bc1; miL[l * 128 + lane * 4 + 1] = bi1;
        mcL[l * 128 + lane * 4 + 2] = bc2; miL[l * 128 + lane * 4 + 2] = bi2;
        mcL[l * 128 + lane * 4 + 3] = bc3; miL[l * 128 + lane * 4 + 3] = bi3;
        __syncthreads();
        if (lane == 0) {                       // lexicographic 4-pass select (stable top_k)
            for (int j = 0; j < 4; ++j) {
                float best = 1e30f; int bidx = 0x7fffffff; int bpos = -1;
                for (int p = 0; p < 128; ++p) {
                    float c = mcL[l * 128 + p]; int i2 = miL[l * 128 + p];
                    if (i2 >= 0 && (c < best || (c == best && i2 < bidx))) {
                        best = c; bidx = i2; bpos = p;
                    }
                }
                rtL[l * 4 + j] = bidx;
                if (bpos >= 0) miL[l * 128 + bpos] = -1;
            }
        }
        __syncthreads();
    }

    // ---- 5. per-pred argmin over selected rows (ascending l, strict <) ----
    if (tid == 0) {
        for (int l = 0; l < MAXL; ++l) {
            if (!gmL[l]) continue;
            for (int j = 0; j < 4; ++j) {
                int   n = rtL[l * 4 + j];
                float c = costL[l * NPRED + n];
                int cur = mgmmL[n];
                if (cur == 0 || c < costL[(cur - 1) * NPRED + n]) mgmmL[n] = l + 1;
            }
        }
    }
    __syncthreads();

    // ---- 6. focal / smooth-L1 / lane-IoU accumulation ----
    float regsum = 0.f, iousum = 0.f, wsum = 0.f;
    const float sc[4] = {71.0f, 799.0f, 180.0f, 71.0f};
    for (int n = tid; n < NPRED; n += 256) {
        const float* p = pred + (size_t)n * DPRED;
        int  mg  = mgmmL[n];
        int  tgt = mg ? 1 : 0;
        float l0 = p[0], l1 = p[1];
        float mx  = fmaxf(l0, l1);
        float lse = mx + logf(expf(l0 - mx) + expf(l1 - mx));
        float lp  = (tgt ? l1 : l0) - lse;
        float pt  = expf(lp);
        float om  = 1.0f - pt;
        float alpha = tgt ? 0.9f : 0.1f;
        cls_part[(size_t)img * NPRED + n] = -alpha * om * om * lp;

        if (mg) {
            int l = mg - 1;
            #pragma unroll
            for (int d = 0; d < 4; ++d) {
                float rp = p[2 + d] * sc[d];
                float rt = gtrL[l * 4 + d] * sc[d];
                float dd = fabsf(rp - rt);
                regsum += (dd < 1.0f) ? 0.5f * dd * dd : dd - 0.5f;
            }
            float ovr = 0.f, uni = 0.f;
            for (int k = 0; k < NPTS; ++k) {
                float rp = p[6 + k] * (799.0f / 800.0f);
                float rt = gtxL[l * NPTS + k];
                float lo = fminf(rp, rt), hi = fmaxf(rp, rt);
                ovr += fmaxf(lo - hi + 2.0f * HALFW, 0.0f);
                uni += hi - lo + 2.0f * HALFW;
            }
            iousum += 1.0f - ovr / fmaxf(uni, 1e-9f);
            wsum   += 1.0f;
        }
    }
    redL[tid] = regsum; redL[256 + tid] = iousum; redL[512 + tid] = wsum;
    __syncthreads();
    if (tid == 0) {                     // fixed-order reduction -> deterministic
        float rs = 0.f, is = 0.f, ws = 0.f;
        for (int i = 0; i < 256; ++i) { rs += redL[i]; is += redL[256 + i]; ws += redL[512 + i]; }
        reg_part[img] = rs / fmaxf(ws * 4.0f, 1.0f);
        iou_part[img] = is / fmaxf(ws, 1.0f);
    }
}

// ===================================================================
// Kernel 2: single block. clsA/clsB means, median via bitonic sort,
// final weighted total.
// ===================================================================
extern "C" __global__ __launch_bounds__(256)
void crit_final_kernel(const float* __restrict__ cls_part,
                       const float* __restrict__ reg_part,
                       const float* __restrict__ iou_part,
                       const float* __restrict__ diff,
                       float* __restrict__ out)
{
    __shared__ float clsA[NPRED];
    __shared__ float clsB[NPRED];
    __shared__ float dv[NPRED];
    __shared__ float red[256];
    __shared__ float s_delta;
    const int tid = threadIdx.x;

    for (int n = tid; n < NPRED; n += 256) {
        float sa = 0.f, sb = 0.f;
        for (int im = 0; im < NIMG; ++im) {
            sa += cls_part[(size_t)im * NPRED + n];
            sb += cls_part[(size_t)(NIMG + im) * NPRED + n];
        }
        sa *= (1.0f / (float)NIMG);
        sb *= (1.0f / (float)NIMG);
        clsA[n] = sa; clsB[n] = sb; dv[n] = sa - sb;
    }
    __syncthreads();

    // bitonic sort dv[0..4095] ascending
    for (int k = 2; k <= NPRED; k <<= 1) {
        for (int j = k >> 1; j > 0; j >>= 1) {
            for (int i = tid; i < NPRED; i += 256) {
                int ixj = i ^ j;
                if (ixj > i) {
                    bool up = ((i & k) == 0);
                    float a = dv[i], c = dv[ixj];
                    if ((up && a > c) || (!up && a < c)) { dv[i] = c; dv[ixj] = a; }
                }
            }
            __syncthreads();
        }
    }
    if (tid == 0) s_delta = 0.5f * (dv[NPRED / 2 - 1] + dv[NPRED / 2]);
    __syncthreads();

    const float h = 0.5f * s_delta;
    float part = 0.f;
    for (int n = tid; n < NPRED; n += 256) {
        float d1 = 0.25f * (diff[n] + diff[NPRED + n] + diff[2 * NPRED + n] + diff[3 * NPRED + n]);
        part += (1.0f - d1) * (clsA[n] - h) + d1 * (clsB[n] + h);
    }
    red[tid] = part;
    __syncthreads();
    if (tid == 0) {
        float cls_loss = 0.f;
        for (int i = 0; i < 256; ++i) cls_loss += red[i];
        float regA = 0.f, regB = 0.f, iouA = 0.f, iouB = 0.f;
        for (int im = 0; im < NIMG; ++im) {
            regA += reg_part[im];        regB += reg_part[NIMG + im];
            iouA += iou_part[im];        iouB += iou_part[NIMG + im];
        }
        const float inv = 1.0f / (float)NIMG;
        regA *= inv; regB *= inv; iouA *= inv; iouB *= inv;
        out[0] = (regA + regB) * 0.5f + (iouA + iouB) * 2.0f + cls_loss * 2.0f;
    }
}

// ===================================================================
extern "C" void kernel_launch(void* const* d_in, const int* in_sizes, int n_in,
                              void* d_out, int out_size, void* d_ws, size_t ws_size,
                              hipStream_t stream) {
    const float* pred_fir = (const float*)d_in[0];
    const float* pred_sec = (const float*)d_in[1];
    const float* gt       = (const float*)d_in[2];
    const float* diff     = (const float*)d_in[3];

    float* cls_part = (float*)d_ws;                       // 2*192*4096 floats
    float* reg_part = cls_part + 2 * NIMG * NPRED;        // 384 floats
    float* iou_part = reg_part + 2 * NIMG;                // 384 floats

    crit_image_kernel<<<2 * NIMG, 256, LDS_TOTAL, stream>>>(
        pred_fir, pred_sec, gt, cls_part, reg_part, iou_part);
    crit_final_kernel<<<1, 256, 0, stream>>>(
        cls_part, reg_part, iou_part, diff, (float*)d_out);
    (void)in_sizes; (void)n_in; (void)out_size; (void)ws_size;
}